// RoIAlignAvg_73117523247234
// MI455X (gfx1250) — compile-verified
//
#include <hip/hip_runtime.h>

// RoIAlign (8x8 grid, bilinear) + avg_pool2x2_s1 -> [N, C, 7, 7]
// MI455X / gfx1250, wave32. Pooling done on the WMMA unit:
//   OUT[c, p] = sum_g G[c, g] * P[g, p],  P constant 64x49 (entries 0 or 0.25)
// via V_WMMA_F32_16X16X4_F32, K-chunked 64 = 16 steps of K=4.
// P is encoded per lane as a 64-bit mask: bit g set iff P[g, p(lane)] = 0.25.

typedef float v2f __attribute__((ext_vector_type(2)));
typedef float v8f __attribute__((ext_vector_type(8)));

#define B_DIM   2
#define C_DIM   256
#define H_DIM   100
#define W_DIM   100
#define GRID_N  8      // 8x8 sample grid
#define POOL_N  7      // 7x7 pooled output
#define OUT_PER_CH (POOL_N * POOL_N)          // 49

__global__ __launch_bounds__(256) void
roialign_avgpool_wmma(const float* __restrict__ feat,
                      const float* __restrict__ rois,
                      const float* __restrict__ scale_p,
                      float* __restrict__ out) {
  // G[c][g] staged in LDS, 2-dword XOR swizzle: logical pair kp at row c lives
  // at physical pair (kp ^ (c & 31)). Conflict-free writes, 8B-aligned reads.
  __shared__ float G[C_DIM * (GRID_N * GRID_N)];   // 64 KB

  const int roi = blockIdx.x;
  const int tid = threadIdx.x;                     // channel in phase 1

  // ---- Phase 1: one bilinear sample per grid point, thread = channel ----
  const float scale = scale_p[0];
  const float* r = rois + (size_t)roi * 5;
  const int   bi = (int)r[0];
  const float x1 = r[1] * scale;
  const float y1 = r[2] * scale;
  const float x2 = r[3] * scale;
  const float y2 = r[4] * scale;
  const float roiw = fmaxf(x2 - x1, 0.0f);
  const float roih = fmaxf(y2 - y1, 0.0f);
  const float binw = roiw / (float)(GRID_N - 1);
  const float binh = roih / (float)(GRID_N - 1);

  const float* fb = feat + ((size_t)bi * C_DIM + tid) * (H_DIM * W_DIM);
  const int csw = tid & 31;

  for (int gy = 0; gy < GRID_N; ++gy) {
    const float ys = y1 + (float)gy * binh;
    const int   vy = (int)(ys >= 0.0f) & (int)(ys < (float)H_DIM);
    int y0 = (int)floorf(ys);
    y0 = min(max(y0, 0), H_DIM - 2);
    const float ly = ys - (float)y0;
    const float wy1v = ly, wy0v = 1.0f - ly;
    const float* frow = fb + y0 * W_DIM;
#pragma unroll
    for (int gx = 0; gx < GRID_N; ++gx) {
      const float xs = x1 + (float)gx * binw;
      const int   v  = vy & (int)(xs >= 0.0f) & (int)(xs < (float)W_DIM);
      int x0 = (int)floorf(xs);
      x0 = min(max(x0, 0), W_DIM - 2);
      const float lx = xs - (float)x0;
      const float wx1v = lx, wx0v = 1.0f - lx;
      const float v00 = frow[x0];
      const float v01 = frow[x0 + 1];
      const float v10 = frow[x0 + W_DIM];
      const float v11 = frow[x0 + W_DIM + 1];
      float val = wy0v * (wx0v * v00 + wx1v * v01)
                + wy1v * (wx0v * v10 + wx1v * v11);
      val = v ? val : 0.0f;
      const int g = gy * GRID_N + gx;
      G[tid * 64 + (((((g >> 1) ^ csw) & 31) << 1) | (g & 1))] = val;
    }
  }
  __syncthreads();

  // ---- Phase 2: pooling as OUT = G * P on the WMMA unit ----
  // job = (cg, t): 16 channel-groups x 4 column tiles = 64 jobs; 8 waves x 8 jobs.
  const int lane = tid & 31;
  const int wid  = tid >> 5;
  const int n16  = lane & 15;     // N within 16x16 D tile
  const int kh   = lane >> 4;     // lane half (selects K offset / M offset)

#pragma unroll 1
  for (int jj = 0; jj < 8; ++jj) {
    const int job = wid * 8 + jj;
    const int cg  = job >> 2;     // channel group (16 channels)
    const int t   = job & 3;      // output-column tile (p = t*16 + n16)
    const int p   = t * 16 + n16;
    const int pv  = (int)(p < OUT_PER_CH);
    const int py  = p / 7;

    // 64-bit pooling mask for this output column p: bits {px,px+1} of rows
    // {py,py+1}, i.e. 0x303 << (px + 8*py) = 0x303 << (p + py). Zero when
    // p >= 49 -> weights all zero, no divergence in the WMMA loop.
    unsigned long long m = 0x303ull << ((unsigned)(p + py) & 63u);
    m = pv ? m : 0ull;
    m >>= (unsigned)(2 * kh);     // lane-half pre-shift: step s, vgpr j -> bit 4s+j
    const unsigned mlo = (unsigned)m;
    const unsigned mhi = (unsigned)(m >> 32);

    const int row = cg * 16 + n16;        // A-matrix row = channel
    const float* arow = &G[row * 64];
    const int rsw = row & 31;

    v8f acc = {0.0f, 0.0f, 0.0f, 0.0f, 0.0f, 0.0f, 0.0f, 0.0f};
#pragma unroll
    for (int s = 0; s < 16; ++s) {
      // A 16x4 f32 layout: VGPR0 = G[row][4s + 2*kh], VGPR1 = G[row][4s + 2*kh + 1]
      const int kp = 2 * s + kh;                         // logical dword pair
      v2f a = *(const v2f*)(arow + (((kp ^ rsw) & 31) << 1));
      // B 4x16 f32 layout: VGPR0 -> K = 2*kh, VGPR1 -> K = 2*kh + 1 (rows of P)
      const unsigned mm = (s < 8) ? mlo : mhi;
      const int sh = (4 * s) & 31;                       // compile-time per step
      const float b0 = ((mm >> (sh + 0)) & 1u) ? 0.25f : 0.0f;
      const float b1 = ((mm >> (sh + 1)) & 1u) ? 0.25f : 0.0f;
      v2f b = { b0, b1 };
      acc = __builtin_amdgcn_wmma_f32_16x16x4_f32(
          /*neg_a=*/false, a, /*neg_b=*/false, b,
          /*c_mod=*/(short)0, acc, /*reuse_a=*/false, /*reuse_b=*/false);
    }

    // D 16x16 f32: VGPR r -> channel (cg*16 + r + 8*kh), column N = n16
    if (pv) {
      const size_t base =
          ((size_t)roi * C_DIM + (size_t)(cg * 16 + 8 * kh)) * OUT_PER_CH + p;
#pragma unroll
      for (int rr = 0; rr < 8; ++rr)
        out[base + (size_t)rr * OUT_PER_CH] = acc[rr];
    }
  }
}

extern "C" void kernel_launch(void* const* d_in, const int* in_sizes, int n_in,
                              void* d_out, int out_size, void* d_ws, size_t ws_size,
                              hipStream_t stream) {
  const float* feat    = (const float*)d_in[0];   // [2,256,100,100] fp32
  const float* rois    = (const float*)d_in[1];   // [N,5] fp32
  const float* scale_p = (const float*)d_in[2];   // scalar fp32
  float* out = (float*)d_out;                     // [N,256,7,7] fp32
  const int n_rois = in_sizes[1] / 5;             // 2048
  roialign_avgpool_wmma<<<dim3(n_rois), dim3(256), 0, stream>>>(feat, rois, scale_p, out);
}